// PointWarping_47373489274954
// MI455X (gfx1250) — compile-verified
//
#include <hip/hip_runtime.h>
#include <math.h>

typedef __attribute__((ext_vector_type(2))) float v2f;
typedef __attribute__((ext_vector_type(8))) float v8f;

#define NPTS 8192
#define NTILES (NPTS / 16)
#define IDXMASK 0x1FFFu          // 13 bits: candidate index 0..8191
#define KEYMASK 0xFFFFE000u      // top 19 bits of f32 distance

// ---------------------------------------------------------------------------
// Prep: pack candidates p = xyz1 + flow1 as float4 { -2x, -2y, -2z, |p|^2 }
// so the WMMA A-matrix rows load with a single b64 per lane, and the exact
// distance to any candidate can be recomputed from one b128 in the epilogue.
// ---------------------------------------------------------------------------
__global__ void pw_prep_kernel(const float* __restrict__ xyz1,
                               const float* __restrict__ flow1,
                               float* __restrict__ cand, int nb) {
    int i = blockIdx.x * blockDim.x + threadIdx.x;
    int total = nb * NPTS;
    if (i >= total) return;
    int b = i / NPTS;
    int p = i - b * NPTS;
    const float* x = xyz1 + (size_t)b * 3 * NPTS;
    const float* f = flow1 + (size_t)b * 3 * NPTS;
    float px = x[p] + f[p];
    float py = x[NPTS + p] + f[NPTS + p];
    float pz = x[2 * NPTS + p] + f[2 * NPTS + p];
    float4 o;
    o.x = -2.0f * px;
    o.y = -2.0f * py;
    o.z = -2.0f * pz;
    o.w = px * px + py * py + pz * pz;
    reinterpret_cast<float4*>(cand)[i] = o;
}

static __device__ __forceinline__ int imin32(int a, int b) { return a < b ? a : b; }
static __device__ __forceinline__ int imax32(int a, int b) { return a < b ? b : a; }

// Streaming insert of one packed key into sorted top-3 (k0 <= k1 <= k2):
// pure v_min_i32 / v_max_i32 network, 5 VALU. Signed compare means a key
// with the f32 sign bit set (tiny negative d^2 from rounding) sorts first,
// i.e. is treated as nearest -- the correct behavior for a ~0 distance.
static __device__ __forceinline__ void kinsert(int k, int& k0, int& k1, int& k2) {
    int t0 = imin32(k, k0);
    int b0 = imax32(k, k0);
    int t1 = imin32(b0, k1);
    int b1 = imax32(b0, k1);
    int t2 = imin32(b1, k2);
    k0 = t0; k1 = t1; k2 = t2;
}

// ---------------------------------------------------------------------------
// Main: one wave32 handles 16 query points. Squared distances to candidate
// tiles are produced 16x16 at a time with v_wmma_f32_16x16x4_f32:
//   D[m][n] = A[m]·B[n] + C = (-2c·q) + |c|^2 + |q|^2
// Each distance is packed with its candidate index into a sortable i32 key
// (upper 19 bits of the f32 | 13-bit index); ties break toward the lower
// index, matching top_k stability. The tile loop is unrolled by 2 with both
// A-operand loads issued up front so the second load's latency hides behind
// the first tile's select network. Top-3 merged across wave halves with one
// shfl_xor(16) per key; exact distances recomputed in the epilogue.
// ---------------------------------------------------------------------------
__global__ void pw_knn_warp_kernel(const float* __restrict__ xyz2,
                                   const float* __restrict__ flow1,
                                   const float* __restrict__ cand,
                                   float* __restrict__ out, int nb) {
    const int lane = threadIdx.x & 31;
    const int lh   = lane >> 4;        // which half of the wave
    const int n    = lane & 15;        // query column within tile
    const int wavesPerBlock = blockDim.x >> 5;
    const int wave = blockIdx.x * wavesPerBlock + (threadIdx.x >> 5);
    const int wavesPerBatch = NPTS / 16;
    const int b  = wave / wavesPerBatch;
    const int qt = wave - b * wavesPerBatch;
    if (b >= nb) return;               // wave-uniform
    const int q = qt * 16 + n;

    // Query coords and fixed B / C operands
    const float* x2 = xyz2 + (size_t)b * 3 * NPTS;
    const float qx = x2[q];
    const float qy = x2[NPTS + q];
    const float qz = x2[2 * NPTS + q];

    v2f Bv;
    if (lh == 0) { Bv.x = qx; Bv.y = qy;   }   // K=0,1
    else         { Bv.x = qz; Bv.y = 1.0f; }   // K=2,3
    const float qsq = qx * qx + qy * qy + qz * qz;
    v8f Cv;
#pragma unroll
    for (int r = 0; r < 8; ++r) Cv[r] = qsq;

    int k0 = 0x7FFFFFFF, k1 = 0x7FFFFFFF, k2 = 0x7FFFFFFF;

    // Per-lane A-row pointer: lane supplies candidate (t*16 + n), loading
    // (-2x,-2y) in the low half-wave and (-2z,|c|^2) in the high half-wave.
    const float* cb = cand + (size_t)b * NPTS * 4 + (size_t)n * 4 + lh * 2;
    const int lh8 = lh * 8;

    for (int t = 0; t < NTILES; t += 2) {
        // Issue both tile loads before consuming either.
        v2f Av0 = *reinterpret_cast<const v2f*>(cb + (size_t)t * 64);
        v2f Av1 = *reinterpret_cast<const v2f*>(cb + (size_t)t * 64 + 64);

        v8f D0 = __builtin_amdgcn_wmma_f32_16x16x4_f32(
            false, Av0, false, Bv, (short)0, Cv, false, false);
        {
            const unsigned base = (unsigned)(t * 16 + lh8);   // multiple of 8
#pragma unroll
            for (int r = 0; r < 8; ++r) {
                int key = (int)((__float_as_uint(D0[r]) & KEYMASK) | (base | (unsigned)r));
                kinsert(key, k0, k1, k2);
            }
        }

        v8f D1 = __builtin_amdgcn_wmma_f32_16x16x4_f32(
            false, Av1, false, Bv, (short)0, Cv, false, false);
        {
            const unsigned base = (unsigned)((t + 1) * 16 + lh8);
#pragma unroll
            for (int r = 0; r < 8; ++r) {
                int key = (int)((__float_as_uint(D1[r]) & KEYMASK) | (base | (unsigned)r));
                kinsert(key, k0, k1, k2);
            }
        }
    }

    // Merge the two half-wave top-3 lists (lane n <-> lane n+16)
    {
        int o0 = __shfl_xor(k0, 16, 32);
        int o1 = __shfl_xor(k1, 16, 32);
        int o2 = __shfl_xor(k2, 16, 32);
        kinsert(o0, k0, k1, k2);
        kinsert(o1, k0, k1, k2);
        kinsert(o2, k0, k1, k2);
    }

    // Lanes 0-15 finish their query: exact distances, IDW weights, warp.
    if (lh == 0) {
        const int j0 = (int)((unsigned)k0 & IDXMASK);
        const int j1 = (int)((unsigned)k1 & IDXMASK);
        const int j2 = (int)((unsigned)k2 & IDXMASK);

        const float4* c4 = reinterpret_cast<const float4*>(cand) + (size_t)b * NPTS;
        float4 c0 = c4[j0], c1 = c4[j1], c2 = c4[j2];
        // c holds (-2x,-2y,-2z,|c|^2): dsq = |c|^2 + |q|^2 - 2 c·q
        float dsq0 = fmaf(c0.x, qx, fmaf(c0.y, qy, fmaf(c0.z, qz, c0.w + qsq)));
        float dsq1 = fmaf(c1.x, qx, fmaf(c1.y, qy, fmaf(c1.z, qz, c1.w + qsq)));
        float dsq2 = fmaf(c2.x, qx, fmaf(c2.y, qy, fmaf(c2.z, qz, c2.w + qsq)));

        float s0 = fmaxf(sqrtf(fmaxf(dsq0, 0.0f)), 1e-10f);
        float s1 = fmaxf(sqrtf(fmaxf(dsq1, 0.0f)), 1e-10f);
        float s2 = fmaxf(sqrtf(fmaxf(dsq2, 0.0f)), 1e-10f);
        float v0 = 1.0f / s0;
        float v1 = 1.0f / s1;
        float v2 = 1.0f / s2;
        float inv = 1.0f / (v0 + v1 + v2);
        float w0 = v0 * inv, w1 = v1 * inv, w2 = v2 * inv;

        const float* fl = flow1 + (size_t)b * 3 * NPTS;
        float fx = w0 * fl[j0]            + w1 * fl[j1]            + w2 * fl[j2];
        float fy = w0 * fl[NPTS + j0]     + w1 * fl[NPTS + j1]     + w2 * fl[NPTS + j2];
        float fz = w0 * fl[2 * NPTS + j0] + w1 * fl[2 * NPTS + j1] + w2 * fl[2 * NPTS + j2];

        float* ob = out + (size_t)b * 3 * NPTS;
        ob[q]            = qx - fx;
        ob[NPTS + q]     = qy - fy;
        ob[2 * NPTS + q] = qz - fz;
    }
}

extern "C" void kernel_launch(void* const* d_in, const int* in_sizes, int n_in,
                              void* d_out, int out_size, void* d_ws, size_t ws_size,
                              hipStream_t stream) {
    const float* xyz1  = (const float*)d_in[0];
    const float* xyz2  = (const float*)d_in[1];
    const float* flow1 = (const float*)d_in[2];
    float* out = (float*)d_out;

    const int nb = in_sizes[0] / (3 * NPTS);   // batch count (2)
    float* cand = (float*)d_ws;                // nb * NPTS * 4 floats (256 KB)

    // Pack candidate rows for the WMMA A-matrix.
    {
        int total = nb * NPTS;
        int threads = 256;
        int blocks = (total + threads - 1) / threads;
        pw_prep_kernel<<<blocks, threads, 0, stream>>>(xyz1, flow1, cand, nb);
    }

    // KNN + warp: one wave per 16 queries, 8 waves per block.
    {
        int totalWaves = nb * (NPTS / 16);     // 1024
        int threads = 256;                     // 8 waves
        int wavesPerBlock = threads / 32;
        int blocks = (totalWaves + wavesPerBlock - 1) / wavesPerBlock;
        pw_knn_warp_kernel<<<blocks, threads, 0, stream>>>(xyz2, flow1, cand, out, nb);
    }
}